// nh_Block_mix_67826123538911
// MI455X (gfx1250) — compile-verified
//
#include <hip/hip_runtime.h>

typedef float v2f __attribute__((ext_vector_type(2)));
typedef float v8f __attribute__((ext_vector_type(8)));

#define BATCH 4
#define T     4096
#define NHB   16
#define MD    128
#define BT    (BATCH * T)          // 16384
#define ROWS  (BT * NHB)           // 262144
#define LN100 4.6051701859880913680f

__device__ __forceinline__ float leaky(float x) { return x >= 0.f ? x : 0.2f * x; }

__device__ __forceinline__ v8f wmma_f32(v2f a, v2f b, v8f c) {
  // V_WMMA_F32_16X16X4_F32: D = A(16x4) x B(4x16) + C, fp32 matrix path
  return __builtin_amdgcn_wmma_f32_16x16x4_f32(false, a, false, b, (short)0, c, false, false);
}

// ---------------------------------------------------------------------------
// Kernel 1: brute-force kNN (top-16 by squared distance, stable like top_k)
// grid (T/256, BATCH), block 256. Writes global row indices (b*T + idx).
// ---------------------------------------------------------------------------
__global__ __launch_bounds__(256) void knn_kernel(const float* __restrict__ coords,
                                                  int* __restrict__ rowmap) {
  __shared__ float2 cc[256];
  const int b = blockIdx.y;
  const int q = blockIdx.x * 256 + threadIdx.x;
  const float2* cb = ((const float2*)coords) + (size_t)b * T;
  const float qx = cb[q].x, qy = cb[q].y;

  float dist[NHB];
  int   idxs[NHB];
#pragma unroll
  for (int i = 0; i < NHB; ++i) { dist[i] = 3.4e38f; idxs[i] = 0x7fffffff; }

  for (int tile = 0; tile < T / 256; ++tile) {
    __syncthreads();
    cc[threadIdx.x] = cb[tile * 256 + threadIdx.x];
    __syncthreads();
    for (int j = 0; j < 256; ++j) {
      const float dx = qx - cc[j].x, dy = qy - cc[j].y;
      const float d = dx * dx + dy * dy;
      const int ci = tile * 256 + j;
      if (d < dist[NHB - 1]) {           // strict: ties keep earlier index (stable)
#pragma unroll
        for (int s = NHB - 1; s > 0; --s) {
          const bool gt = dist[s - 1] > d;       // shift element if strictly greater
          const float dj = gt ? dist[s - 1] : d;
          const int   ij = gt ? idxs[s - 1] : ci;
          if (dist[s] > d) { dist[s] = dj; idxs[s] = ij; }
        }
        if (dist[0] > d) { dist[0] = d; idxs[0] = ci; }
      }
    }
  }
#pragma unroll
  for (int s = 0; s < NHB; ++s)
    rowmap[((size_t)b * T + q) * NHB + s] = b * T + idxs[s];
}

// ---------------------------------------------------------------------------
// Kernel 2: h = LN1( x[nbr] + PE(rel coords) ), one wave per output row
// ---------------------------------------------------------------------------
__global__ __launch_bounds__(256) void build_h_kernel(
    const float* __restrict__ x, const float* __restrict__ coords,
    const int* __restrict__ rowmap,
    const float* __restrict__ pe_w1, const float* __restrict__ pe_b1,
    const float* __restrict__ pe_w2, const float* __restrict__ pe_b2,
    const float* __restrict__ g, const float* __restrict__ bvec,
    float* __restrict__ hout) {
  __shared__ float hid[8][128];
  const int lane = threadIdx.x & 31, w = threadIdx.x >> 5;
  const long r = (long)blockIdx.x * 8 + w;     // row in [0, ROWS)
  const long bt = r >> 4;                      // center token (global row into x)
  const int nbr = rowmap[r];
  const float csx = coords[2 * nbr]     - coords[2 * bt];
  const float csy = coords[2 * nbr + 1] - coords[2 * bt + 1];

#pragma unroll
  for (int i = 0; i < 4; ++i) {
    const int col = lane + 32 * i;
    hid[w][col] = leaky(csx * pe_w1[col] + csy * pe_w1[128 + col] + pe_b1[col]);
  }
  __syncthreads();

  float vals[4];
#pragma unroll
  for (int i = 0; i < 4; ++i) {
    const int col = lane + 32 * i;
    float acc = pe_b2[col];
    for (int k = 0; k < 128; ++k) acc += hid[w][k] * pe_w2[k * 128 + col];
    vals[i] = x[(long)nbr * MD + col] + acc;
  }
  float s = 0.f, ss = 0.f;
#pragma unroll
  for (int i = 0; i < 4; ++i) { s += vals[i]; ss += vals[i] * vals[i]; }
#pragma unroll
  for (int m = 1; m < 32; m <<= 1) { s += __shfl_xor(s, m, 32); ss += __shfl_xor(ss, m, 32); }
  const float mean = s * (1.f / 128.f);
  const float inv = rsqrtf(ss * (1.f / 128.f) - mean * mean + 1e-5f);
#pragma unroll
  for (int i = 0; i < 4; ++i) {
    const int col = lane + 32 * i;
    hout[r * MD + col] = (vals[i] - mean) * inv * g[col] + bvec[col];
  }
}

// ---------------------------------------------------------------------------
// Generic fp32 WMMA GEMM: C[M,N] = act(gather(A)[M,K] @ B[K,N] + bias)
// Block tile 128M x 64N, 8 waves of 32x32, K staged 32-wide in LDS.
// Requires M%128==0, N%64==0, K%32==0 (true for all call sites).
// ---------------------------------------------------------------------------
__global__ __launch_bounds__(256) void gemm_kernel(
    const float* __restrict__ A, const int* __restrict__ rowmap,
    const float* __restrict__ B, const float* __restrict__ bias,
    float* __restrict__ C, int M, int N, int K, int act) {
  __shared__ float ldsA[128 * 34];   // [row][k], stride 34 (pad, 8B-aligned float2)
  __shared__ float ldsB[64 * 34];    // [col][k] transposed, stride 34

  const int tid = threadIdx.x;
  const int lane = tid & 31;
  const int w = tid >> 5;
  const int mOff = (w & 3) * 32;
  const int nOff = (w >> 2) * 32;
  const long rowBase = (long)blockIdx.y * 128;
  const int colBase = blockIdx.x * 64;

  // cooperative-load indices
  const int arow = tid >> 1;             // 0..127
  const int acol = (tid & 1) * 16;       // 0 or 16
  const long garow = rowBase + arow;
  const float* aptr = A + (rowmap ? (long)rowmap[garow] : garow) * (long)K;
  const int brow = tid >> 3;             // 0..31
  const int bcol = (tid & 7) * 8;        // 0..56

  const int koff = (lane >> 4) * 2;
  const int msel = lane & 15;

  v8f acc[2][2] = {};

  for (int k0 = 0; k0 < K; k0 += 32) {
    // A tile: 128x32
    {
      const float4* src = (const float4*)(aptr + k0 + acol);
      const float4 a0 = src[0], a1 = src[1], a2 = src[2], a3 = src[3];
      float2* dst2 = (float2*)&ldsA[arow * 34 + acol];
      dst2[0] = make_float2(a0.x, a0.y); dst2[1] = make_float2(a0.z, a0.w);
      dst2[2] = make_float2(a1.x, a1.y); dst2[3] = make_float2(a1.z, a1.w);
      dst2[4] = make_float2(a2.x, a2.y); dst2[5] = make_float2(a2.z, a2.w);
      dst2[6] = make_float2(a3.x, a3.y); dst2[7] = make_float2(a3.z, a3.w);
    }
    // B tile: 32x64, stored transposed
    {
      const float4* src = (const float4*)(B + (long)(k0 + brow) * N + colBase + bcol);
      const float4 b0 = src[0], b1 = src[1];
      ldsB[(bcol + 0) * 34 + brow] = b0.x; ldsB[(bcol + 1) * 34 + brow] = b0.y;
      ldsB[(bcol + 2) * 34 + brow] = b0.z; ldsB[(bcol + 3) * 34 + brow] = b0.w;
      ldsB[(bcol + 4) * 34 + brow] = b1.x; ldsB[(bcol + 5) * 34 + brow] = b1.y;
      ldsB[(bcol + 6) * 34 + brow] = b1.z; ldsB[(bcol + 7) * 34 + brow] = b1.w;
    }
    if (k0 + 32 < K) {   // prefetch next tiles into cache (global_prefetch)
      __builtin_prefetch((const void*)(aptr + k0 + 32 + acol), 0, 3);
      __builtin_prefetch((const void*)(B + (long)(k0 + 32 + brow) * N + colBase + bcol), 0, 3);
    }
    __syncthreads();
#pragma unroll
    for (int kk = 0; kk < 32; kk += 4) {
      const v2f a0 = *(const v2f*)&ldsA[(mOff + msel) * 34 + kk + koff];
      const v2f a1 = *(const v2f*)&ldsA[(mOff + 16 + msel) * 34 + kk + koff];
      const v2f b0 = *(const v2f*)&ldsB[(nOff + msel) * 34 + kk + koff];
      const v2f b1 = *(const v2f*)&ldsB[(nOff + 16 + msel) * 34 + kk + koff];
      acc[0][0] = wmma_f32(a0, b0, acc[0][0]);
      acc[0][1] = wmma_f32(a0, b1, acc[0][1]);
      acc[1][0] = wmma_f32(a1, b0, acc[1][0]);
      acc[1][1] = wmma_f32(a1, b1, acc[1][1]);
    }
    __syncthreads();
  }

#pragma unroll
  for (int ms = 0; ms < 2; ++ms)
#pragma unroll
    for (int ns = 0; ns < 2; ++ns) {
      const int col = colBase + nOff + ns * 16 + msel;
      const float bb = bias ? bias[col] : 0.f;
#pragma unroll
      for (int rr = 0; rr < 8; ++rr) {
        const long row = rowBase + mOff + ms * 16 + rr + ((lane >> 4) << 3);
        float v = acc[ms][ns][rr] + bb;
        if (act) v = leaky(v);
        C[row * (long)N + col] = v;
      }
    }
}

// ---------------------------------------------------------------------------
// Kernel 4: cosine multi-head attention over the 16-neighborhood.
// One thread per (token, head, query-neighbor). Writes over its own Q slice.
// ---------------------------------------------------------------------------
__global__ __launch_bounds__(256) void attn_kernel(float* __restrict__ Q,
                                                   const float* __restrict__ Kb,
                                                   const float* __restrict__ V,
                                                   const float* __restrict__ logit_scale) {
  const long gid = (long)blockIdx.x * 256 + threadIdx.x;
  const int n = (int)(gid & 15);
  const int head = (int)((gid >> 4) & 3);
  const long bt = gid >> 6;

  float* qp = Q + (bt * NHB + n) * MD + head * 32;
  float4 qv[8];
  float qn = 0.f;
#pragma unroll
  for (int i = 0; i < 8; ++i) {
    qv[i] = ((const float4*)qp)[i];
    qn += qv[i].x * qv[i].x + qv[i].y * qv[i].y + qv[i].z * qv[i].z + qv[i].w * qv[i].w;
  }
  const float invq = 1.f / fmaxf(sqrtf(qn), 1e-6f);
  const float scale = expf(fminf(logit_scale[head], LN100));

  float logits[16];
#pragma unroll
  for (int m = 0; m < 16; ++m) {
    const float4* kp = (const float4*)(Kb + (bt * NHB + m) * MD + head * 32);
    float dot = 0.f, kn = 0.f;
#pragma unroll
    for (int i = 0; i < 8; ++i) {
      const float4 kv = kp[i];
      dot += qv[i].x * kv.x + qv[i].y * kv.y + qv[i].z * kv.z + qv[i].w * kv.w;
      kn  += kv.x * kv.x + kv.y * kv.y + kv.z * kv.z + kv.w * kv.w;
    }
    logits[m] = dot * invq / fmaxf(sqrtf(kn), 1e-6f) * scale;
  }
  float mx = logits[0];
#pragma unroll
  for (int m = 1; m < 16; ++m) mx = fmaxf(mx, logits[m]);
  float se = 0.f;
#pragma unroll
  for (int m = 0; m < 16; ++m) { logits[m] = expf(logits[m] - mx); se += logits[m]; }
  const float inv = 1.f / se;

  float4 out[8];
#pragma unroll
  for (int i = 0; i < 8; ++i) out[i] = make_float4(0.f, 0.f, 0.f, 0.f);
#pragma unroll
  for (int m = 0; m < 16; ++m) {
    const float wgt = logits[m] * inv;
    const float4* vp = (const float4*)(V + (bt * NHB + m) * MD + head * 32);
#pragma unroll
    for (int i = 0; i < 8; ++i) {
      const float4 vv = vp[i];
      out[i].x += wgt * vv.x; out[i].y += wgt * vv.y;
      out[i].z += wgt * vv.z; out[i].w += wgt * vv.w;
    }
  }
  float4* op = (float4*)qp;
#pragma unroll
  for (int i = 0; i < 8; ++i) op[i] = out[i];
}

// ---------------------------------------------------------------------------
// Kernel 5: h2 = LN2(h + att_proj), wave per 128-row (safe in-place)
// ---------------------------------------------------------------------------
__global__ __launch_bounds__(256) void ln2_kernel(const float* h, const float* att,
                                                  const float* __restrict__ g,
                                                  const float* __restrict__ bvec,
                                                  float* out) {
  const int lane = threadIdx.x & 31, w = threadIdx.x >> 5;
  const long r = (long)blockIdx.x * 8 + w;
  float vals[4];
  float s = 0.f, ss = 0.f;
#pragma unroll
  for (int i = 0; i < 4; ++i) {
    const int col = lane + 32 * i;
    const float v = h[r * MD + col] + att[r * MD + col];
    vals[i] = v; s += v; ss += v * v;
  }
#pragma unroll
  for (int m = 1; m < 32; m <<= 1) { s += __shfl_xor(s, m, 32); ss += __shfl_xor(ss, m, 32); }
  const float mean = s * (1.f / 128.f);
  const float inv = rsqrtf(ss * (1.f / 128.f) - mean * mean + 1e-5f);
#pragma unroll
  for (int i = 0; i < 4; ++i) {
    const int col = lane + 32 * i;
    out[r * MD + col] = (vals[i] - mean) * inv * g[col] + bvec[col];
  }
}

// ---------------------------------------------------------------------------
// Kernel 6: fused per-neighbor MLP  h += leaky(leaky(h@w1+b1)@w2+b2)
// 4 waves/block, each wave owns a 16-row tile; WMMA over 16-col chunks of w1.
// ---------------------------------------------------------------------------
__global__ __launch_bounds__(128) void mlp_nh_kernel(float* h,
                                                     const float* __restrict__ w1,
                                                     const float* __restrict__ b1,
                                                     const float* __restrict__ w2,
                                                     const float* __restrict__ b2) {
  __shared__ float tile[4][16 * 130];
  __shared__ float mrow[4][16];
  const int lane = threadIdx.x & 31, w = threadIdx.x >> 5;
  const long t0 = ((long)blockIdx.x * 4 + w) * 16;
  float* src = h + t0 * MD;

  for (int e = lane; e < 16 * MD; e += 32)
    tile[w][(e >> 7) * 130 + (e & 127)] = src[e];
  __syncthreads();

  const int msel = lane & 15, koff = (lane >> 4) * 2;
  v8f msum = {};
  for (int nc = 0; nc < 32; ++nc) {
    v8f acc = {};
    const int col = nc * 16 + msel;
#pragma unroll
    for (int kk = 0; kk < 128; kk += 4) {
      const v2f a = *(const v2f*)&tile[w][msel * 130 + kk + koff];
      v2f bb;
      bb.x = w1[(kk + koff) * 512 + col];
      bb.y = w1[(kk + koff + 1) * 512 + col];
      acc = wmma_f32(a, bb, acc);
    }
    const float bias1 = b1[col];
    const float w2v = w2[col];
#pragma unroll
    for (int rr = 0; rr < 8; ++rr) msum[rr] += leaky(acc[rr] + bias1) * w2v;
  }
#pragma unroll
  for (int rr = 0; rr < 8; ++rr) {
    float v = msum[rr];
    v += __shfl_xor(v, 1, 32); v += __shfl_xor(v, 2, 32);
    v += __shfl_xor(v, 4, 32); v += __shfl_xor(v, 8, 32);
    msum[rr] = leaky(v + b2[0]);
  }
  if (lane == 0) {
#pragma unroll
    for (int rr = 0; rr < 8; ++rr) mrow[w][rr] = msum[rr];
  }
  if (lane == 16) {
#pragma unroll
    for (int rr = 0; rr < 8; ++rr) mrow[w][8 + rr] = msum[rr];
  }
  __syncthreads();
  for (int e = lane; e < 16 * MD; e += 32)
    src[e] = tile[w][(e >> 7) * 130 + (e & 127)] + mrow[w][e >> 7];
}

// ---------------------------------------------------------------------------
// Kernel 7: LN3 over unfolded rows of 2048, one block per row
// ---------------------------------------------------------------------------
__global__ __launch_bounds__(256) void ln3_kernel(const float* __restrict__ in,
                                                  const float* __restrict__ g,
                                                  const float* __restrict__ bvec,
                                                  float* __restrict__ out) {
  __shared__ float redS[8], redQ[8];
  const long r = blockIdx.x;
  float vals[8];
  float s = 0.f, ss = 0.f;
#pragma unroll
  for (int i = 0; i < 8; ++i) {
    const int col = threadIdx.x + 256 * i;
    const float v = in[r * 2048 + col];
    vals[i] = v; s += v; ss += v * v;
  }
#pragma unroll
  for (int m = 1; m < 32; m <<= 1) { s += __shfl_xor(s, m, 32); ss += __shfl_xor(ss, m, 32); }
  const int lane = threadIdx.x & 31, w = threadIdx.x >> 5;
  if (lane == 0) { redS[w] = s; redQ[w] = ss; }
  __syncthreads();
  s = 0.f; ss = 0.f;
#pragma unroll
  for (int i = 0; i < 8; ++i) { s += redS[i]; ss += redQ[i]; }
  const float mean = s * (1.f / 2048.f);
  const float inv = rsqrtf(ss * (1.f / 2048.f) - mean * mean + 1e-5f);
#pragma unroll
  for (int i = 0; i < 8; ++i) {
    const int col = threadIdx.x + 256 * i;
    out[r * 2048 + col] = (vals[i] - mean) * inv * g[col] + bvec[col];
  }
}

// ---------------------------------------------------------------------------
extern "C" void kernel_launch(void* const* d_in, const int* in_sizes, int n_in,
                              void* d_out, int out_size, void* d_ws, size_t ws_size,
                              hipStream_t stream) {
  const float* x      = (const float*)d_in[0];
  const float* coords = (const float*)d_in[1];
  const float* w_v    = (const float*)d_in[2];
  const float* w_q    = (const float*)d_in[3];
  const float* w_k    = (const float*)d_in[4];
  const float* pe_w1  = (const float*)d_in[5];
  const float* pe_b1  = (const float*)d_in[6];
  const float* pe_w2  = (const float*)d_in[7];
  const float* pe_b2  = (const float*)d_in[8];
  const float* lscale = (const float*)d_in[9];
  const float* w_o    = (const float*)d_in[10];
  const float* b_o    = (const float*)d_in[11];
  const float* nh_w1  = (const float*)d_in[12];
  const float* nh_b1  = (const float*)d_in[13];
  const float* nh_w2  = (const float*)d_in[14];
  const float* nh_b2  = (const float*)d_in[15];
  const float* u_w1   = (const float*)d_in[16];
  const float* u_b1   = (const float*)d_in[17];
  const float* u_w2   = (const float*)d_in[18];
  const float* u_b2   = (const float*)d_in[19];
  const float* ln1_g  = (const float*)d_in[20];
  const float* ln1_b  = (const float*)d_in[21];
  const float* ln2_g  = (const float*)d_in[22];
  const float* ln2_b  = (const float*)d_in[23];
  const float* ln3_g  = (const float*)d_in[24];
  const float* ln3_b  = (const float*)d_in[25];

  const size_t BUF = (size_t)ROWS * MD;          // 33,554,432 floats
  float* bufA = (float*)d_ws;                    // h / h2 / h3
  float* bufQ = bufA + BUF;                      // q / attn-out / ln3-out
  float* bufK = bufQ + BUF;                      // k / u1
  float* bufV = bufK + BUF;                      // v / att_proj
  int* rowmap = (int*)(bufV + BUF);              // ROWS ints

  // 1. kNN -> rowmap
  knn_kernel<<<dim3(T / 256, BATCH), 256, 0, stream>>>(coords, rowmap);
  // 2. h = LN1(gather(x) + PE)
  build_h_kernel<<<ROWS / 8, 256, 0, stream>>>(x, coords, rowmap, pe_w1, pe_b1,
                                               pe_w2, pe_b2, ln1_g, ln1_b, bufA);
  // 3-5. q, k, v projections (v gathers raw x rows via rowmap)
  gemm_kernel<<<dim3(2, ROWS / 128), 256, 0, stream>>>(bufA, nullptr, w_q, nullptr, bufQ, ROWS, MD, MD, 0);
  gemm_kernel<<<dim3(2, ROWS / 128), 256, 0, stream>>>(bufA, nullptr, w_k, nullptr, bufK, ROWS, MD, MD, 0);
  gemm_kernel<<<dim3(2, ROWS / 128), 256, 0, stream>>>(x, rowmap, w_v, nullptr, bufV, ROWS, MD, MD, 0);
  // 6. attention (writes over own Q slice)
  attn_kernel<<<(BT * 64) / 256, 256, 0, stream>>>(bufQ, bufK, bufV, lscale);
  // 7. output projection
  gemm_kernel<<<dim3(2, ROWS / 128), 256, 0, stream>>>(bufQ, nullptr, w_o, b_o, bufV, ROWS, MD, MD, 0);
  // 8. h2 = LN2(h + att)
  ln2_kernel<<<ROWS / 8, 256, 0, stream>>>(bufA, bufV, ln2_g, ln2_b, bufA);
  // 9. h3 = h2 + leaky(leaky(h2@w1+b1)@w2+b2)   (in place)
  mlp_nh_kernel<<<ROWS / 64, 128, 0, stream>>>(bufA, nh_w1, nh_b1, nh_w2, nh_b2);
  // 10. LN3 over unfolded [BT, 2048]
  ln3_kernel<<<BT, 256, 0, stream>>>(bufA, ln3_g, ln3_b, bufQ);
  // 11-12. big mixes: [16384,2048]@[2048,2048] x2 with fused leaky
  gemm_kernel<<<dim3(32, BT / 128), 256, 0, stream>>>(bufQ, nullptr, u_w1, u_b1, bufK, BT, 2048, 2048, 1);
  gemm_kernel<<<dim3(32, BT / 128), 256, 0, stream>>>(bufK, nullptr, u_w2, u_b2, (float*)d_out, BT, 2048, 2048, 1);
}